// voltreservoir_kernel_layer_87875030876959
// MI455X (gfx1250) — compile-verified
//
#include <hip/hip_runtime.h>
#include <hip/hip_bf16.h>

typedef float v2f __attribute__((ext_vector_type(2)));
typedef float v8f __attribute__((ext_vector_type(8)));

#define NPTS  2048
#define DFEAT 48
#define KCH   16          // DFEAT / 3 chunks
#define TILES 128         // NPTS / 16
#define WAVES 8           // waves (tiles) per block

// One wave32 computes one 16x16 output tile via 16x V_WMMA_F32_16X16X4_F32
// (one per 3-wide feature chunk, K padded 3->4 with a zero slot in LDS).
__global__ __launch_bounds__(256) void volt_wmma_kernel(
    const float* __restrict__ vct,     // (2048, 48)
    const float* __restrict__ posnum,  // [tau, lambda]
    float* __restrict__ out)           // (2048, 2048)
{
  // 4-padded chunk rows: [wave][side][row][chunk*4 + d], slot d==3 is 0.
  __shared__ __align__(16) float rowBuf[WAVES][2][16][4 * KCH];
  // per-chunk squared norms: [wave][side][chunk][row]
  __shared__ __align__(16) float sqTab[WAVES][2][KCH][16];

  const int wave = threadIdx.x >> 5;
  const int lane = threadIdx.x & 31;
  const int hi   = lane >> 4;      // lane half: K={0,1} vs K={2,pad}
  const int col  = lane & 15;

  const int tile = blockIdx.x * WAVES + wave;
  const int ti = tile >> 7;            // output row-tile
  const int tj = tile & (TILES - 1);   // output col-tile

  // ---- Phase 1: read vct rows once, write padded rows + squared norms to LDS
  {
    const int side = hi;              // lanes 0-15: A rows (ti), 16-31: B rows (tj)
    const int grow = (side ? tj : ti) * 16 + col;
    const float4* p4 = reinterpret_cast<const float4*>(vct + grow * DFEAT);
    float r[DFEAT];
#pragma unroll
    for (int i = 0; i < DFEAT / 4; ++i) {
      float4 q = p4[i];
      r[4*i+0] = q.x; r[4*i+1] = q.y; r[4*i+2] = q.z; r[4*i+3] = q.w;
    }
    float4* dst = reinterpret_cast<float4*>(&rowBuf[wave][side][col][0]);
#pragma unroll
    for (int k = 0; k < KCH; ++k) {
      float x = r[3*k], y = r[3*k+1], z = r[3*k+2];
      dst[k] = float4{x, y, z, 0.0f};               // ds_store_b128, pad = 0
      sqTab[wave][side][k][col] = x*x + y*y + z*z;
    }
  }
  __syncthreads();

  const float tau  = posnum[0];
  const float lam  = posnum[1];
  const float tau2 = tau * tau;
  const float lam2 = lam * lam;
  const float c1   = 1.0f - tau2;   // denom = c1 + tau2 * dist

  // Fragment base pointers in LDS (A-frag lane holds M=col; B-frag N=col).
  const float* rA = &rowBuf[wave][0][col][0];
  const float* rB = &rowBuf[wave][1][col][0];

  float cp[8], acc[8];
#pragma unroll
  for (int v = 0; v < 8; ++v) { cp[v] = 1.0f; acc[v] = 0.0f; }
  float w = lam2;  // lambda^{2k}, k = 1..16

#pragma unroll
  for (int k = 0; k < KCH; ++k) {
    // A 16x4 fp32: lanes<16 -> dims {0,1}; lanes>=16 -> dims {2, 0}. B mirrors.
    // Single aligned ds_load_b64 each; pad slot in LDS is zero -> branch-free.
    v2f a = *reinterpret_cast<const v2f*>(rA + 4*k + 2*hi);
    v2f b = *reinterpret_cast<const v2f*>(rB + 4*k + 2*hi);

    v8f c = {0.f, 0.f, 0.f, 0.f, 0.f, 0.f, 0.f, 0.f};
    c = __builtin_amdgcn_wmma_f32_16x16x4_f32(false, a, false, b,
                                              (short)0, c, false, false);

    // broadcast reads: sa[v] = |A row (v+8*hi)|^2, sb = |B row col|^2
    float4 q0 = *reinterpret_cast<const float4*>(&sqTab[wave][0][k][hi * 8]);
    float4 q1 = *reinterpret_cast<const float4*>(&sqTab[wave][0][k][hi * 8 + 4]);
    float  sb = sqTab[wave][1][k][col];
    float sa[8] = {q0.x, q0.y, q0.z, q0.w, q1.x, q1.y, q1.z, q1.w};

#pragma unroll
    for (int v = 0; v < 8; ++v) {
      // d2 and dist; raw v_sqrt_f32 / v_rcp_f32 (operands are benign:
      // d2 in [0, ~1e3], denom in [c1, c1 + tau2*dist] well away from 0).
      float d2   = fmaxf(fmaf(-2.0f, c[v], sa[v] + sb), 0.0f);
      float dist = __builtin_amdgcn_sqrtf(d2);        // sqrt(0)=0, matches ref mask
      float inv  = __builtin_amdgcn_rcpf(fmaf(tau2, dist, c1));
      cp[v]  *= inv;                                   // running cumprod over chunks
      acc[v]  = fmaf(w, cp[v], acc[v]);                // weighted prefix-product sum
    }
    w *= lam2;
  }

  // C/D layout: VGPR v <-> output row (v + 8*hi), N = col. Coalesced 64B rows;
  // output is write-once -> non-temporal stores.
  float* obase = out + (ti * 16 + 8 * hi) * NPTS + tj * 16 + col;
#pragma unroll
  for (int v = 0; v < 8; ++v) {
    __builtin_nontemporal_store(1.0f + acc[v], obase + v * NPTS);
  }
}

extern "C" void kernel_launch(void* const* d_in, const int* in_sizes, int n_in,
                              void* d_out, int out_size, void* d_ws, size_t ws_size,
                              hipStream_t stream) {
  const float* vct    = (const float*)d_in[0];
  const float* posnum = (const float*)d_in[1];
  float* out          = (float*)d_out;

  const int total_tiles = TILES * TILES;          // 16384
  const int blocks      = total_tiles / WAVES;    // 2048
  volt_wmma_kernel<<<blocks, 256, 0, stream>>>(vct, posnum, out);
}